// GCN_32160715112815
// MI455X (gfx1250) — compile-verified
//
#include <hip/hip_runtime.h>
#include <hip/hip_bf16.h>
#include <math.h>

// GCN 3-layer forward for MI455X (gfx1250, wave32).
//  - GEMMs: V_WMMA_F32_16X16X4_F32 (full fp32 matrix pipe), pre-packed
//    transposed/padded B -> inner loop is pure b64 loads + 16 chained WMMAs.
//  - ReLU fused into next layer's A-fragment loads (template param).
//  - Bias fused into aggregation-buffer init (replaces the zero pass).
//  - Edge aggregation: float4 gathers + global_atomic_add_f32 scatters
//    (the bandwidth-dominant part: ~1.2 GB -> ~55us at 23.3 TB/s).

#define GN   100000   // nodes
#define GE   800000   // edges
#define LDH  64       // feature stride (D_IN == D_HID == 64; D_OUT padded to 64)
#define DOUT 47

typedef __attribute__((ext_vector_type(2))) float v2f;
typedef __attribute__((ext_vector_type(8))) float v8f;

// ---------------------------------------------------------------------------
// Pack B [64 x Ncols] into Bt [64 x 64]: Bt[n*64 + k] = B[k*Ncols + n],
// zero for n >= Ncols. Makes the GEMM B-fragment a contiguous v2f load.
// ---------------------------------------------------------------------------
__global__ __launch_bounds__(256) void gcn_pack_bt(
    const float* __restrict__ B, float* __restrict__ Bt, int Ncols)
{
    int t = blockIdx.x * blockDim.x + threadIdx.x;
    if (t >= 64 * 64) return;
    int n = t >> 6;
    int k = t & 63;
    Bt[t] = (n < Ncols) ? B[k * Ncols + n] : 0.0f;
}

// Pad bias [Ncols] to bp [64] (zeros beyond Ncols).
__global__ __launch_bounds__(64) void gcn_pad_bias(
    const float* __restrict__ b, float* __restrict__ bp, int Ncols)
{
    int c = threadIdx.x;
    bp[c] = (c < Ncols) ? b[c] : 0.0f;
}

// agg[n][0:64] = bp[0:64]  (bias-broadcast init; replaces zeroing)
__global__ __launch_bounds__(256) void gcn_init_agg(
    float4* __restrict__ agg, const float4* __restrict__ bp4, int n4)
{
    int t = blockIdx.x * blockDim.x + threadIdx.x;
    if (t < n4) agg[t] = bp4[t & 15];
}

// ---------------------------------------------------------------------------
// C[M x 64] = act(A)[M x 64] @ Bt^T, act = ReLU if RELU_A (fused on A loads).
// One wave per 16x16 tile; K=64 -> 16 chained f32 WMMA ops.
// A fragment (32-bit 16x4): lanes 0-15 hold K=k,k+1 ; lanes 16-31 K=k+2,k+3.
// B fragment mirrors it via the transposed pack -> identical addressing.
// ---------------------------------------------------------------------------
template <bool RELU_A>
__global__ __launch_bounds__(256) void gcn_gemm_wmma_f32(
    const float* __restrict__ A, const float* __restrict__ Bt,
    float* __restrict__ C, int M)
{
    const int lane    = threadIdx.x & 31;
    const int wave    = (blockIdx.x * blockDim.x + threadIdx.x) >> 5;
    const int rowTile = wave >> 2;        // 4 col-tiles of 16 cover 64-wide C
    const int colTile = wave & 3;
    if (rowTile * 16 >= M) return;

    const int mrow = rowTile * 16 + (lane & 15);
    const int kh   = (lane >> 4) << 1;                  // 0 or 2
    const int n    = colTile * 16 + (lane & 15);

    const float* aBase = A  + ((size_t)mrow << 6) + kh;
    const float* bBase = Bt + ((size_t)n    << 6) + kh;

    v8f acc = {};
#pragma unroll
    for (int k = 0; k < 64; k += 4) {
        v2f a = *(const v2f*)(aBase + k);               // global_load_b64
        v2f b = *(const v2f*)(bBase + k);               // global_load_b64
        if (RELU_A) {
            a.x = fmaxf(a.x, 0.0f);
            a.y = fmaxf(a.y, 0.0f);
        }
        acc = __builtin_amdgcn_wmma_f32_16x16x4_f32(
            /*neg_a=*/false, a, /*neg_b=*/false, b,
            /*c_mod=*/(short)0, acc, /*reuse_a=*/false, /*reuse_b=*/false);
    }

    // D layout: VGPR i -> row (rowTile*16 + i + 8*(lane>=16)), col n.
    const int rbase = rowTile * 16 + ((lane >> 4) << 3);
    float* cBase = C + ((size_t)rbase << 6) + n;
#pragma unroll
    for (int i = 0; i < 8; ++i)
        cBase[(size_t)i << 6] = acc[i];
}

// ---------------------------------------------------------------------------
// agg[dst[e], :] += h[src[e], :]  -- 16 threads per edge, float4 per thread.
__global__ __launch_bounds__(256) void gcn_scatter_add(
    const float* __restrict__ h, float* __restrict__ agg,
    const int* __restrict__ src, const int* __restrict__ dst, int E)
{
    int t = blockIdx.x * blockDim.x + threadIdx.x;
    if (t >= E * 16) return;
    const int e  = t >> 4;
    const int c4 = (t & 15) << 2;
    const int s  = src[e];
    const int d  = dst[e];
    const float4 v = *(const float4*)(h + ((size_t)s << 6) + c4);  // global_load_b128
    float* a = agg + ((size_t)d << 6) + c4;
    atomicAdd(a + 0, v.x);   // global_atomic_add_f32 (no return)
    atomicAdd(a + 1, v.y);
    atomicAdd(a + 2, v.z);
    atomicAdd(a + 3, v.w);
}

// Per-node log_softmax over 47 classes (bias already folded into agg).
// One wave per node, wave32 shuffle reductions.
__global__ __launch_bounds__(256) void gcn_log_softmax(
    const float* __restrict__ agg, float* __restrict__ out, int N)
{
    const int wave = (blockIdx.x * blockDim.x + threadIdx.x) >> 5;
    const int lane = threadIdx.x & 31;
    if (wave >= N) return;
    const float* row = agg + ((size_t)wave << 6);
    const float NEG_INF = -__builtin_inff();
    float v0 = (lane      < DOUT) ? row[lane]      : NEG_INF;
    float v1 = (lane + 32 < DOUT) ? row[lane + 32] : NEG_INF;

    float m = fmaxf(v0, v1);
#pragma unroll
    for (int off = 16; off > 0; off >>= 1)
        m = fmaxf(m, __shfl_xor(m, off, 32));

    float s = ((lane      < DOUT) ? expf(v0 - m) : 0.0f)
            + ((lane + 32 < DOUT) ? expf(v1 - m) : 0.0f);
#pragma unroll
    for (int off = 16; off > 0; off >>= 1)
        s += __shfl_xor(s, off, 32);

    const float ls = logf(s) + m;
    if (lane      < DOUT) out[(size_t)wave * DOUT + lane]      = v0 - ls;
    if (lane + 32 < DOUT) out[(size_t)wave * DOUT + lane + 32] = v1 - ls;
}

// ---------------------------------------------------------------------------
extern "C" void kernel_launch(void* const* d_in, const int* in_sizes, int n_in,
                              void* d_out, int out_size, void* d_ws, size_t ws_size,
                              hipStream_t stream)
{
    const float* x  = (const float*)d_in[0];
    const int*   ei = (const int*)  d_in[1];   // [2, E]: row 0 = src, row 1 = dst
    const float* W1 = (const float*)d_in[2];
    const float* b1 = (const float*)d_in[3];
    const float* W2 = (const float*)d_in[4];
    const float* b2 = (const float*)d_in[5];
    const float* W3 = (const float*)d_in[6];
    const float* b3 = (const float*)d_in[7];
    float* out = (float*)d_out;

    const int* src = ei;
    const int* dst = ei + GE;

    float* ws0 = (float*)d_ws;                   // h   : N x 64 fp32
    float* ws1 = ws0 + (size_t)GN * LDH;         // agg : N x 64 fp32
    float* bt  = ws1 + (size_t)GN * LDH;         // 64 x 64 packed weights
    float* bp  = bt + 64 * 64;                   // 64 padded bias

    const dim3 blk(256);
    const int gemmBlocks = ((GN / 16) * 4) / 8;  // 1 wave / 16x16 tile
    const int packBlocks = (64 * 64) / 256;
    const int initBlocks = (GN * 16 + 255) / 256;
    const int scatBlocks = (GE * 16 + 255) / 256;
    const int smaxBlocks = (GN + 7) / 8;         // 8 waves / block

    // ---- Layer 1: h = x@W1 ; agg = b1 + scatter(h) ----
    gcn_pack_bt             <<<packBlocks, blk, 0, stream>>>(W1, bt, 64);
    gcn_pad_bias            <<<1, 64, 0, stream>>>(b1, bp, 64);
    gcn_gemm_wmma_f32<false><<<gemmBlocks, blk, 0, stream>>>(x, bt, ws0, GN);
    gcn_init_agg            <<<initBlocks, blk, 0, stream>>>((float4*)ws1, (const float4*)bp, GN * 16);
    gcn_scatter_add         <<<scatBlocks, blk, 0, stream>>>(ws0, ws1, src, dst, GE);

    // ---- Layer 2: h = relu(agg)@W2 ; agg = b2 + scatter(h) ----
    gcn_pack_bt             <<<packBlocks, blk, 0, stream>>>(W2, bt, 64);
    gcn_pad_bias            <<<1, 64, 0, stream>>>(b2, bp, 64);
    gcn_gemm_wmma_f32<true> <<<gemmBlocks, blk, 0, stream>>>(ws1, bt, ws0, GN);
    gcn_init_agg            <<<initBlocks, blk, 0, stream>>>((float4*)ws1, (const float4*)bp, GN * 16);
    gcn_scatter_add         <<<scatBlocks, blk, 0, stream>>>(ws0, ws1, src, dst, GE);

    // ---- Layer 3 (47 classes, zero-padded to 64) + log_softmax ----
    gcn_pack_bt             <<<packBlocks, blk, 0, stream>>>(W3, bt, DOUT);
    gcn_pad_bias            <<<1, 64, 0, stream>>>(b3, bp, DOUT);
    gcn_gemm_wmma_f32<true> <<<gemmBlocks, blk, 0, stream>>>(ws1, bt, ws0, GN);
    gcn_init_agg            <<<initBlocks, blk, 0, stream>>>((float4*)ws1, (const float4*)bp, GN * 16);
    gcn_scatter_add         <<<scatBlocks, blk, 0, stream>>>(ws0, ws1, src, dst, GE);
    gcn_log_softmax         <<<smaxBlocks, blk, 0, stream>>>(ws1, out, GN);
}